// Threshold_weights3_52699248721945
// MI455X (gfx1250) — compile-verified
//
#include <hip/hip_runtime.h>
#include <math.h>

typedef __attribute__((ext_vector_type(4))) float f32x4;
typedef __attribute__((ext_vector_type(2))) float f32x2;
typedef __attribute__((ext_vector_type(8))) float f32x8;

#define C_DIM 1000
#define B_DIM 16384
#define NPAIRS (B_DIM * 4)          // (row, matrix) pairs
#define GRID_B 2048
#define WPB 8                       // waves per block (256 threads, wave32)

__device__ __forceinline__ void top2_update(float v, float& t1, float& t2) {
    if (v > t1) { t2 = t1; t1 = v; }
    else if (v > t2) { t2 = v; }
}

// ---------------- Kernel B: per-(row,matrix) top-2 margin + per-block max ----
__global__ __launch_bounds__(256) void margins_kernel(
    const float* __restrict__ o1, const float* __restrict__ o2,
    const float* __restrict__ o3, const float* __restrict__ mim,
    const int* __restrict__ targets,
    float* __restrict__ ws_margins,   // [B*4]
    float* __restrict__ ws_partial)   // [GRID_B]
{
    __shared__ float smax[WPB];
    const int tid  = threadIdx.x;
    const int lane = tid & 31;
    const int wave = tid >> 5;
    const float* mats[4] = { o1, o2, o3, mim };

    const float NEG_INF = -__builtin_inff();
    float wmax = NEG_INF;

    const int gwave = blockIdx.x * WPB + wave;
    // NPAIRS / (GRID_B*WPB) == 4 iterations, uniform across all waves
    for (int p = gwave; p < NPAIRS; p += GRID_B * WPB) {
        const int row = p >> 2;
        const int mat = p & 3;
        const float* rp  = mats[mat] + (size_t)row * C_DIM;
        const f32x4* rp4 = (const f32x4*)rp;        // rows are 16B-aligned (4000B stride)

        float t1 = NEG_INF, t2 = NEG_INF;
        #pragma unroll
        for (int i = 0; i < 8; ++i) {
            int idx = lane + 32 * i;                // 250 float4 per row
            if (idx < 250) {
                f32x4 v = __builtin_nontemporal_load(rp4 + idx);  // NT: stream, don't thrash L2
                top2_update(v.x, t1, t2);
                top2_update(v.y, t1, t2);
                top2_update(v.z, t1, t2);
                top2_update(v.w, t1, t2);
            }
        }
        // Wave-level top-2 merge: top1=max(a1,b1); top2=max(min(a1,b1), max(a2,b2))
        #pragma unroll
        for (int m = 16; m >= 1; m >>= 1) {
            float u1 = __shfl_xor(t1, m, 32);
            float u2 = __shfl_xor(t2, m, 32);
            float n1 = fmaxf(t1, u1);
            float n2 = fmaxf(fminf(t1, u1), fmaxf(t2, u2));
            t1 = n1; t2 = n2;
        }
        if (lane == 0) {
            float tv = rp[targets[row]];
            ws_margins[p] = (tv == t1) ? (t1 - t2) : 0.0f;  // value-equality, like reference
        }
        if (mat != 3) wmax = fmaxf(wmax, t1);       // mimic excluded from max_preds
    }
    if (lane == 0) smax[wave] = wmax;
    __syncthreads();
    if (tid == 0) {
        float bm = smax[0];
        #pragma unroll
        for (int i = 1; i < WPB; ++i) bm = fmaxf(bm, smax[i]);
        ws_partial[blockIdx.x] = bm;
    }
}

// ---------------- Kernel C: reduce 2048 partials -> d_out[0] ----------------
__global__ __launch_bounds__(256) void maxreduce_kernel(
    const float* __restrict__ ws_partial, float* __restrict__ d_out)
{
    __shared__ float s[256];
    float m = -__builtin_inff();
    for (int i = threadIdx.x; i < GRID_B; i += 256)
        m = fmaxf(m, ws_partial[i]);
    s[threadIdx.x] = m;
    __syncthreads();
    #pragma unroll
    for (int off = 128; off > 0; off >>= 1) {
        if (threadIdx.x < off) s[threadIdx.x] = fmaxf(s[threadIdx.x], s[threadIdx.x + off]);
        __syncthreads();
    }
    if (threadIdx.x == 0) d_out[0] = s[0];
}

// ---------------- Kernel D: softmax over 4 margins, WMMA row-sum ------------
// One wave handles 16 rows. A = exp tile (16x4 f32, ISA layout: lane&15 = M,
// lane>>4 selects K-pair), B = ones (4x16, layout-invariant), D[m][n] = row sum.
__global__ __launch_bounds__(256) void softmax_kernel(
    const float* __restrict__ ws_margins, float* __restrict__ out /* = d_out+1 */)
{
    const int tid   = threadIdx.x;
    const int lane  = tid & 31;
    const int wave  = tid >> 5;
    const int gwave = blockIdx.x * WPB + wave;
    const int rowBase = gwave * 16;
    const int r     = lane & 15;    // matrix row M for this lane
    const int khalf = lane >> 4;    // 0 -> K={0,1}, 1 -> K={2,3}

    const f32x2* mp = (const f32x2*)(ws_margins + (size_t)(rowBase + r) * 4 + khalf * 2);
    f32x2 mg = *mp;
    float x0 = mg.x * 0.5f;         // margins / T, T = 2
    float x1 = mg.y * 0.5f;

    // row max across the 4 values: local pair max, exchange with lane^16
    float lmax = fmaxf(x0, x1);
    float rmax = fmaxf(lmax, __shfl_xor(lmax, 16, 32));
    float e0 = __expf(x0 - rmax);
    float e1 = __expf(x1 - rmax);

    // WMMA: D = A(16x4 exps) * B(4x16 ones) + 0  ->  D[m][*] = sum_k exp[m][k]
    f32x2 a; a.x = e0; a.y = e1;
    f32x2 b; b.x = 1.0f; b.y = 1.0f;
    f32x8 acc = {};
    acc = __builtin_amdgcn_wmma_f32_16x16x4_f32(
        /*neg_a=*/false, a, /*neg_b=*/false, b,
        /*c_mod=*/(short)0, acc, /*reuse_a=*/false, /*reuse_b=*/false);

    // D layout: vgpr j, lanes 0-15 -> row j; lanes 16-31 -> row 8+j.
    float own   = acc[r & 7];
    float other = __shfl_xor(own, 16, 32);
    float sum   = ((r >> 3) == khalf) ? own : other;

    float inv = 1.0f / sum;
    size_t o = (size_t)(rowBase + r) * 4 + khalf * 2;
    out[o]     = e0 * inv;
    out[o + 1] = e1 * inv;
}

// ---------------------------------------------------------------------------
extern "C" void kernel_launch(void* const* d_in, const int* in_sizes, int n_in,
                              void* d_out, int out_size, void* d_ws, size_t ws_size,
                              hipStream_t stream) {
    const float* o1  = (const float*)d_in[0];
    const float* o2  = (const float*)d_in[1];
    const float* o3  = (const float*)d_in[2];
    const float* mim = (const float*)d_in[3];
    const int* targets = (const int*)d_in[4];
    float* out = (float*)d_out;

    float* ws_margins = (float*)d_ws;            // B*4 floats = 256 KB
    float* ws_partial = ws_margins + NPAIRS;     // GRID_B floats

    margins_kernel<<<GRID_B, 256, 0, stream>>>(o1, o2, o3, mim, targets,
                                               ws_margins, ws_partial);
    maxreduce_kernel<<<1, 256, 0, stream>>>(ws_partial, out);
    softmax_kernel<<<B_DIM / 16 / WPB, 256, 0, stream>>>(ws_margins, out + 1);
}